// UP_66236985639732
// MI455X (gfx1250) — compile-verified
//
#include <hip/hip_runtime.h>
#include <hip/hip_bf16.h>

typedef float v2f __attribute__((ext_vector_type(2)));
typedef float v8f __attribute__((ext_vector_type(8)));

#define BN 2
#define DIMX 96
#define C2X 48
#define DIX 96
#define NSX 16
#define RX 3
#define KX 4
#define CTX 35          // R + 2*NS
#define H0 64
#define W0 64
#define HHX 128
#define WHX 128
#define LLX (HHX*WHX)   // 16384
#define NCHUNK 128
#define CSZ (LLX/NCHUNK) // 128

__device__ __forceinline__ float sigmoidf_(float x) { return 1.0f / (1.0f + __expf(-x)); }
__device__ __forceinline__ float softplusf_(float x) { return x > 20.0f ? x : log1pf(__expf(x)); }

// position in xc (row-major h*W+w) that feeds xs[k] at scan index t
__device__ __forceinline__ int scan_map(int k, int t) {
    int l = (k >= 2) ? (LLX - 1 - t) : t;
    if (k & 1) { int w = l / HHX; int h = l - w * HHX; return h * WHX + w; }
    return l;
}

// ---------------- K1: upsample 2x bilinear (align-corners style) + affine + pwconv + relu + LN
// writes xb (B,L,C2) row-major (== res, pre-LN) and xn (B,L,C2) (post-LN)
__global__ void k1_up_pw_ln(const float* __restrict__ x,
                            const float* __restrict__ dww, const float* __restrict__ dwb,
                            const float* __restrict__ pww, const float* __restrict__ pwb,
                            const float* __restrict__ lnw, const float* __restrict__ lnb,
                            float* __restrict__ xb, float* __restrict__ xn) {
    __shared__ float v[DIMX];
    __shared__ float sv[C2X];
    int blk = blockIdx.x;
    int b = blk / LLX;
    int pos = blk - b * LLX;
    int h = pos / WHX;
    int w = pos - h * WHX;
    const float sc = 63.0f / 127.0f;
    float ch = h * sc, cw = w * sc;
    int i0 = (int)ch; int i1 = i0 + 1; if (i1 > 63) i1 = 63;
    int j0 = (int)cw; int j1 = j0 + 1; if (j1 > 63) j1 = 63;
    float fh = ch - (float)i0, fw = cw - (float)j0;
    int t = threadIdx.x;
    if (t < DIMX) {
        const float* xc0 = x + ((size_t)(b * DIMX + t) * H0) * W0;
        float a00 = xc0[i0 * W0 + j0], a10 = xc0[i1 * W0 + j0];
        float a01 = xc0[i0 * W0 + j1], a11 = xc0[i1 * W0 + j1];
        float val = (a00 * (1.0f - fh) + a10 * fh) * (1.0f - fw)
                  + (a01 * (1.0f - fh) + a11 * fh) * fw;
        v[t] = val * dww[t] + dwb[t];
    }
    __syncthreads();
    size_t row = (size_t)b * LLX + pos;
    if (t < C2X) {
        float acc = pwb[t];
        const float* wr = pww + (size_t)t * DIMX;
        #pragma unroll 8
        for (int c = 0; c < DIMX; ++c) acc += v[c] * wr[c];
        acc = fmaxf(acc, 0.0f);
        sv[t] = acc;
        xb[row * C2X + t] = acc;
    }
    __syncthreads();
    if (t < C2X) {
        float m = 0.0f;
        for (int c = 0; c < C2X; ++c) m += sv[c];
        m *= (1.0f / C2X);
        float var = 0.0f;
        for (int c = 0; c < C2X; ++c) { float d = sv[c] - m; var += d * d; }
        var *= (1.0f / C2X);
        xn[row * C2X + t] = (sv[t] - m) * rsqrtf(var + 1e-5f) * lnw[t] + lnb[t];
    }
}

// ---------------- K2: in_proj GEMM (32768x48)@(48x192)+b via fp32 WMMA 16x16x4
// xin, z stored channel-major (B,DI,L)
__global__ void k2_inproj_wmma(const float* __restrict__ xn,
                               const float* __restrict__ W, const float* __restrict__ bias,
                               float* __restrict__ xin, float* __restrict__ z) {
    int lane = threadIdx.x & 31, wave = threadIdx.x >> 5;
    int tile = blockIdx.x * 8 + wave;              // 24576 tiles: 2048 M x 12 N
    int mt = tile / 12, nt = tile - mt * 12;
    int M0 = mt * 16, N0 = nt * 16;
    int half = lane >> 4, lr = lane & 15;
    const float* arow = xn + (size_t)(M0 + lr) * C2X;
    v8f acc = {0.f,0.f,0.f,0.f,0.f,0.f,0.f,0.f};
    for (int k = 0; k < C2X; k += 4) {
        v2f a, bm;
        a.x = arow[k + 2 * half];
        a.y = arow[k + 1 + 2 * half];
        bm.x = W[(size_t)(k + 2 * half) * 192 + N0 + lr];
        bm.y = W[(size_t)(k + 1 + 2 * half) * 192 + N0 + lr];
        acc = __builtin_amdgcn_wmma_f32_16x16x4_f32(false, a, false, bm, (short)0, acc, false, false);
    }
    int col = N0 + lr;
    float bv = bias[col];
    #pragma unroll
    for (int r = 0; r < 8; ++r) {
        int row = M0 + r + 8 * half;
        int b = row / LLX; int pos = row - b * LLX;
        float val = acc[r] + bv;
        if (col < DIX) xin[((size_t)b * DIX + col) * LLX + pos] = val;
        else           z  [((size_t)b * DIX + (col - DIX)) * LLX + pos] = val;
    }
}

// ---------------- K3: depthwise 3x3 conv + bias + SiLU (xin -> xc), both (B,DI,H,W)
__global__ void k3_dwconv_silu(const float* __restrict__ xin,
                               const float* __restrict__ cw, const float* __restrict__ cb,
                               float* __restrict__ xc) {
    int idx = blockIdx.x * blockDim.x + threadIdx.x;   // B*DI*L
    int pos = idx % LLX;
    int d = (idx / LLX) % DIX;
    int b = idx / (DIX * LLX);
    int h = pos / WHX, w = pos - h * WHX;
    const float* src = xin + ((size_t)b * DIX + d) * LLX;
    const float* wt = cw + (size_t)d * 9;
    float acc = cb[d];
    #pragma unroll
    for (int kh = 0; kh < 3; ++kh) {
        int ih = h + kh - 1;
        if (ih < 0 || ih >= HHX) continue;
        #pragma unroll
        for (int kw = 0; kw < 3; ++kw) {
            int iw = w + kw - 1;
            if (iw < 0 || iw >= WHX) continue;
            acc += wt[kh * 3 + kw] * src[ih * WHX + iw];
        }
    }
    xc[((size_t)b * DIX + d) * LLX + pos] = acc * sigmoidf_(acc);
}

// ---------------- K4: x_proj per (b,k): (L x 96)@(96 x 35) via WMMA (N padded to 48)
// xd stored position-major: xd[(bk*L + l)*35 + c]
__global__ void k4_xproj_wmma(const float* __restrict__ xc, const float* __restrict__ Wp,
                              float* __restrict__ xd) {
    int lane = threadIdx.x & 31, wave = threadIdx.x >> 5;
    int tile = blockIdx.x * 8 + wave;              // 24576: 8 bk x 1024 M x 3 N
    int bk = tile / 3072; int rem = tile - bk * 3072;
    int mt = rem / 3, nt = rem - mt * 3;
    int b = bk >> 2, k = bk & 3;
    int M0 = mt * 16, N0 = nt * 16;
    int half = lane >> 4, lr = lane & 15;
    int mrow = scan_map(k, M0 + lr);               // xs row -> xc position
    const float* xcb = xc + (size_t)b * DIX * LLX;
    const float* wk = Wp + (size_t)k * CTX * DIX;  // (35,96) row-major; B[d][c] = wk[c*96+d]
    int c = N0 + lr;
    v8f acc = {0.f,0.f,0.f,0.f,0.f,0.f,0.f,0.f};
    for (int kk = 0; kk < DIX; kk += 4) {
        v2f a, bm;
        a.x = xcb[(size_t)(kk + 2 * half) * LLX + mrow];
        a.y = xcb[(size_t)(kk + 1 + 2 * half) * LLX + mrow];
        bm.x = (c < CTX) ? wk[(size_t)c * DIX + kk + 2 * half] : 0.0f;
        bm.y = (c < CTX) ? wk[(size_t)c * DIX + kk + 1 + 2 * half] : 0.0f;
        acc = __builtin_amdgcn_wmma_f32_16x16x4_f32(false, a, false, bm, (short)0, acc, false, false);
    }
    if (c < CTX) {
        #pragma unroll
        for (int r = 0; r < 8; ++r) {
            int l = M0 + r + 8 * half;
            xd[((size_t)bk * LLX + l) * CTX + c] = acc[r];
        }
    }
}

// ---------------- K5a: chunked scan phase A: per-chunk transfer function (P = prod dA, Q = h from 0)
__global__ void k5a_scan_chunks(const float* __restrict__ xd, const float* __restrict__ xc,
                                const float* __restrict__ dtw, const float* __restrict__ dtb,
                                const float* __restrict__ Alogs,
                                float* __restrict__ P, float* __restrict__ Q) {
    int blk = blockIdx.x;              // BN*KX*NCHUNK
    int chunk = blk % NCHUNK;
    int bk = blk / NCHUNK;
    int b = bk >> 2, k = bk & 3;
    int d = threadIdx.x;
    if (d >= DIX) return;
    float Ad[NSX];
    #pragma unroll
    for (int n = 0; n < NSX; ++n) Ad[n] = -__expf(Alogs[((size_t)(k * DIX) + d) * NSX + n]);
    float w0 = dtw[((size_t)(k * DIX) + d) * RX + 0];
    float w1 = dtw[((size_t)(k * DIX) + d) * RX + 1];
    float w2 = dtw[((size_t)(k * DIX) + d) * RX + 2];
    float bb = dtb[k * DIX + d];
    const float* xdb = xd + (size_t)bk * LLX * CTX;
    const float* xcb = xc + ((size_t)b * DIX + d) * LLX;
    float h[NSX], p[NSX];
    #pragma unroll
    for (int n = 0; n < NSX; ++n) { h[n] = 0.0f; p[n] = 1.0f; }
    int t0 = chunk * CSZ;
    for (int tt = 0; tt < CSZ; ++tt) {
        int t = t0 + tt;
        const float* row = xdb + (size_t)t * CTX;
        float delta = softplusf_(bb + w0 * row[0] + w1 * row[1] + w2 * row[2]);
        float du = delta * xcb[scan_map(k, t)];
        #pragma unroll
        for (int n = 0; n < NSX; ++n) {
            float dA = __expf(delta * Ad[n]);
            h[n] = dA * h[n] + du * row[3 + n];
            p[n] *= dA;
        }
    }
    size_t base = (((size_t)bk * NCHUNK + chunk) * DIX + d) * NSX;
    #pragma unroll
    for (int n = 0; n < NSX; ++n) { P[base + n] = p[n]; Q[base + n] = h[n]; }
}

// ---------------- K5b: phase B: chain chunk states; Q[c] becomes h at chunk start
__global__ void k5b_chain(float* __restrict__ Q, const float* __restrict__ P) {
    int tid = blockIdx.x * blockDim.x + threadIdx.x;     // BN*KX*DIX*NSX = 12288
    if (tid >= BN * KX * DIX * NSX) return;
    int bk = tid / (DIX * NSX);
    int dn = tid - bk * (DIX * NSX);
    float hh = 0.0f;
    for (int c = 0; c < NCHUNK; ++c) {
        size_t idx = ((size_t)bk * NCHUNK + c) * (DIX * NSX) + dn;
        float p = P[idx], q = Q[idx];
        Q[idx] = hh;                 // h entering chunk c
        hh = p * hh + q;
    }
}

// ---------------- K5c: phase C: re-run chunks from correct states, emit y + Ds*u
// ys stored (B,K,DI,L)
__global__ void k5c_scan_emit(const float* __restrict__ xd, const float* __restrict__ xc,
                              const float* __restrict__ dtw, const float* __restrict__ dtb,
                              const float* __restrict__ Alogs, const float* __restrict__ Ds,
                              const float* __restrict__ Q, float* __restrict__ ys) {
    int blk = blockIdx.x;
    int chunk = blk % NCHUNK;
    int bk = blk / NCHUNK;
    int b = bk >> 2, k = bk & 3;
    int d = threadIdx.x;
    if (d >= DIX) return;
    float Ad[NSX];
    #pragma unroll
    for (int n = 0; n < NSX; ++n) Ad[n] = -__expf(Alogs[((size_t)(k * DIX) + d) * NSX + n]);
    float w0 = dtw[((size_t)(k * DIX) + d) * RX + 0];
    float w1 = dtw[((size_t)(k * DIX) + d) * RX + 1];
    float w2 = dtw[((size_t)(k * DIX) + d) * RX + 2];
    float bb = dtb[k * DIX + d];
    float Dv = Ds[k * DIX + d];
    const float* xdb = xd + (size_t)bk * LLX * CTX;
    const float* xcb = xc + ((size_t)b * DIX + d) * LLX;
    float* ysb = ys + ((size_t)bk * DIX + d) * LLX;
    float h[NSX];
    size_t base = (((size_t)bk * NCHUNK + chunk) * DIX + d) * NSX;
    #pragma unroll
    for (int n = 0; n < NSX; ++n) h[n] = Q[base + n];
    int t0 = chunk * CSZ;
    for (int tt = 0; tt < CSZ; ++tt) {
        int t = t0 + tt;
        const float* row = xdb + (size_t)t * CTX;
        float delta = softplusf_(bb + w0 * row[0] + w1 * row[1] + w2 * row[2]);
        float u = xcb[scan_map(k, t)];
        float du = delta * u;
        float y = 0.0f;
        #pragma unroll
        for (int n = 0; n < NSX; ++n) {
            float dA = __expf(delta * Ad[n]);
            h[n] = dA * h[n] + du * row[3 + n];
            y += h[n] * row[3 + NSX + n];
        }
        ysb[t] = y + Dv * u;
    }
}

// ---------------- K6: merge 4 directions + out_norm LN + silu(z) gate -> g (B,L,DI) row-major
__global__ void k6_combine_ln_gate(const float* __restrict__ ys, const float* __restrict__ z,
                                   const float* __restrict__ onw, const float* __restrict__ onb,
                                   float* __restrict__ g) {
    __shared__ float yv[DIX];
    int blk = blockIdx.x;
    int b = blk / LLX;
    int pos = blk - b * LLX;
    int h = pos / WHX, w = pos - h * WHX;
    int pt = w * HHX + h;
    int d = threadIdx.x;
    if (d < DIX) {
        size_t s0 = ((size_t)(b * KX + 0) * DIX + d) * LLX;
        size_t s1 = ((size_t)(b * KX + 1) * DIX + d) * LLX;
        size_t s2 = ((size_t)(b * KX + 2) * DIX + d) * LLX;
        size_t s3 = ((size_t)(b * KX + 3) * DIX + d) * LLX;
        yv[d] = ys[s0 + pos] + ys[s2 + (LLX - 1 - pos)]
              + ys[s1 + pt]  + ys[s3 + (LLX - 1 - pt)];
    }
    __syncthreads();
    if (d < DIX) {
        float m = 0.0f;
        for (int c = 0; c < DIX; ++c) m += yv[c];
        m *= (1.0f / DIX);
        float var = 0.0f;
        for (int c = 0; c < DIX; ++c) { float t = yv[c] - m; var += t * t; }
        var *= (1.0f / DIX);
        float yn = (yv[d] - m) * rsqrtf(var + 1e-5f) * onw[d] + onb[d];
        float zv = z[((size_t)b * DIX + d) * LLX + pos];
        g[((size_t)b * LLX + pos) * DIX + d] = yn * (zv * sigmoidf_(zv));
    }
}

// ---------------- K7: out_proj GEMM (32768x96)@(96x48)+b+res via WMMA; out (B,C2,H,W)
__global__ void k7_outproj_wmma(const float* __restrict__ g, const float* __restrict__ W,
                                const float* __restrict__ bias, const float* __restrict__ res,
                                float* __restrict__ out) {
    int lane = threadIdx.x & 31, wave = threadIdx.x >> 5;
    int tile = blockIdx.x * 8 + wave;              // 6144: 2048 M x 3 N
    int mt = tile / 3, nt = tile - mt * 3;
    int M0 = mt * 16, N0 = nt * 16;
    int half = lane >> 4, lr = lane & 15;
    const float* arow = g + (size_t)(M0 + lr) * DIX;
    v8f acc = {0.f,0.f,0.f,0.f,0.f,0.f,0.f,0.f};
    for (int k = 0; k < DIX; k += 4) {
        v2f a, bm;
        a.x = arow[k + 2 * half];
        a.y = arow[k + 1 + 2 * half];
        bm.x = W[(size_t)(k + 2 * half) * C2X + N0 + lr];
        bm.y = W[(size_t)(k + 1 + 2 * half) * C2X + N0 + lr];
        acc = __builtin_amdgcn_wmma_f32_16x16x4_f32(false, a, false, bm, (short)0, acc, false, false);
    }
    int col = N0 + lr;
    float bv = bias[col];
    #pragma unroll
    for (int r = 0; r < 8; ++r) {
        int row = M0 + r + 8 * half;
        int b = row / LLX; int pos = row - b * LLX;
        out[((size_t)b * C2X + col) * LLX + pos] = acc[r] + bv + res[(size_t)row * C2X + col];
    }
}

extern "C" void kernel_launch(void* const* d_in, const int* in_sizes, int n_in,
                              void* d_out, int out_size, void* d_ws, size_t ws_size,
                              hipStream_t stream) {
    const float* x    = (const float*)d_in[0];
    const float* dww  = (const float*)d_in[1];
    const float* dwb  = (const float*)d_in[2];
    const float* pww  = (const float*)d_in[3];
    const float* pwb  = (const float*)d_in[4];
    const float* lnw  = (const float*)d_in[5];
    const float* lnb  = (const float*)d_in[6];
    const float* ipw  = (const float*)d_in[7];
    const float* ipb  = (const float*)d_in[8];
    const float* cw   = (const float*)d_in[9];
    const float* cb   = (const float*)d_in[10];
    const float* xpw  = (const float*)d_in[11];
    const float* dtw  = (const float*)d_in[12];
    const float* dtb  = (const float*)d_in[13];
    const float* alog = (const float*)d_in[14];
    const float* Dsv  = (const float*)d_in[15];
    const float* onw  = (const float*)d_in[16];
    const float* onb  = (const float*)d_in[17];
    const float* opw  = (const float*)d_in[18];
    const float* opb  = (const float*)d_in[19];
    float* out = (float*)d_out;

    float* ws = (float*)d_ws;
    size_t o = 0;
    float* xb  = ws + o; o += (size_t)BN * LLX * C2X;            // 1.57M
    float* xn  = ws + o; o += (size_t)BN * LLX * C2X;            // 1.57M
    float* xin = ws + o; o += (size_t)BN * DIX * LLX;            // 3.15M
    float* zb  = ws + o; o += (size_t)BN * DIX * LLX;            // 3.15M
    float* xc  = ws + o; o += (size_t)BN * DIX * LLX;            // 3.15M
    float* xd  = ws + o; o += (size_t)BN * KX * LLX * CTX;       // 4.59M
    float* Pb  = ws + o; o += (size_t)BN * KX * NCHUNK * DIX * NSX; // 1.57M
    float* Qb  = ws + o; o += (size_t)BN * KX * NCHUNK * DIX * NSX; // 1.57M
    float* ysb = ws + o; o += (size_t)BN * KX * DIX * LLX;       // 12.58M
    float* gb  = ws + o; o += (size_t)BN * LLX * DIX;            // 3.15M
    (void)o; (void)ws_size; (void)in_sizes; (void)n_in; (void)out_size;

    k1_up_pw_ln<<<BN * LLX, 128, 0, stream>>>(x, dww, dwb, pww, pwb, lnw, lnb, xb, xn);
    k2_inproj_wmma<<<3072, 256, 0, stream>>>(xn, ipw, ipb, xin, zb);
    k3_dwconv_silu<<<(BN * DIX * LLX) / 256, 256, 0, stream>>>(xin, cw, cb, xc);
    k4_xproj_wmma<<<3072, 256, 0, stream>>>(xc, xpw, xd);
    k5a_scan_chunks<<<BN * KX * NCHUNK, 128, 0, stream>>>(xd, xc, dtw, dtb, alog, Pb, Qb);
    k5b_chain<<<48, 256, 0, stream>>>(Qb, Pb);
    k5c_scan_emit<<<BN * KX * NCHUNK, 128, 0, stream>>>(xd, xc, dtw, dtb, alog, Dsv, Qb, ysb);
    k6_combine_ln_gate<<<BN * LLX, 128, 0, stream>>>(ysb, zb, onw, onb, gb);
    k7_outproj_wmma<<<768, 256, 0, stream>>>(gb, opw, opb, xb, out);
}